// SREL_56367150792892
// MI455X (gfx1250) — compile-verified
//
#include <hip/hip_runtime.h>
#include <hip/hip_bf16.h>

// MI455X / gfx1250: fp32 WMMA pipeline with algebraic K-split + async-LDS staging.
//
// out[b,l] = phi[b,l] - ( (sum_m relu(ts[b] + tw[b,m] + tv[m])) @ W2 + 16*b2 )[l]
//   ts[b]   = [cos|sin](phi[b])/32 @ W1[0:2048,:]          (WMMA GEMM, 256 rows)
//   tw[b,m] = [wr|wi](b,:,m)       @ W1[2048:4096,:]       (WMMA GEMM, 4096 rows)
//   tv[m]   = v_M[:,m]             @ W1[4096:4608,:] + b1  (tiny VALU GEMM)
// GEMM2 runs on the m-reduced 256-row matrix (linearity of W2).

typedef __attribute__((ext_vector_type(2))) float v2f;
typedef __attribute__((ext_vector_type(8))) float v8f;
typedef __attribute__((ext_vector_type(4))) int   v4i;

#define BB 256
#define LS 1024
#define HH 4096
#define KS 2048     // K of the s-part GEMM
#define KW 2048     // K of the w-part GEMM

#define TM 128      // block tile rows
#define TN 64       // block tile cols
#define KC 32       // k chunk
#define ASTR (KC + 4)   // LDS A stride (16B-aligned rows)
#define BSTR (TN + 4)   // LDS B stride (16B-aligned rows)

#if defined(__gfx1250__) && __has_builtin(__builtin_amdgcn_global_load_async_to_lds_b128)
#define HAVE_ASYNC 1
#else
#define HAVE_ASYNC 0
#endif

__device__ __forceinline__ void wait_async_zero() {
#if HAVE_ASYNC
#if __has_builtin(__builtin_amdgcn_s_wait_asynccnt)
    __builtin_amdgcn_s_wait_asynccnt(0);
#else
    asm volatile("s_wait_asynccnt 0" ::: "memory");
#endif
#endif
}

// Copy 16B from global to LDS (async DMA if available, else through VGPRs).
__device__ __forceinline__ void cp16_g2lds(const float* __restrict__ gsrc,
                                           float* ldst) {
#if HAVE_ASYNC
    __builtin_amdgcn_global_load_async_to_lds_b128(
        (__attribute__((address_space(1))) v4i*)(v4i*)const_cast<float*>(gsrc),
        (__attribute__((address_space(3))) v4i*)(v4i*)ldst, 0, 0);
#else
    *(float4*)ldst = *(const float4*)gsrc;
#endif
}

// Stage B tile (KC x TN) from row-major W (leading dim ldw).
__device__ __forceinline__ void load_B_tile(const float* __restrict__ W, int ldw,
                                            int k0, int Cb, int tid,
                                            float (*Bs)[BSTR]) {
    #pragma unroll
    for (int e = tid; e < (KC * TN) / 4; e += 256) {
        int kk = e >> 4;
        int j4 = e & 15;
        cp16_g2lds(W + (size_t)(k0 + kk) * ldw + Cb + j4 * 4, &Bs[kk][j4 * 4]);
    }
}

// Stage dense A tile (TM x KC) from row-major A (leading dim lda).
__device__ __forceinline__ void load_A_dense(const float* __restrict__ A, int lda,
                                             int Rb, int k0, int tid,
                                             float (*As)[ASTR]) {
    #pragma unroll
    for (int e = tid; e < (TM * KC) / 4; e += 256) {
        int i  = e >> 3;
        int k4 = e & 7;
        cp16_g2lds(A + (size_t)(Rb + i) * lda + k0 + k4 * 4, &As[i][k4 * 4]);
    }
}

// Stage gathered A tile for the w-part GEMM: row r=b*16+m, kg<1024 -> wr, else wi.
// Lane mapping: 16 consecutive lanes hit 16 consecutive m's (64B contiguous).
__device__ __forceinline__ void load_A_w(const float* __restrict__ wr,
                                         const float* __restrict__ wi,
                                         int Rb, int k0, int tid,
                                         float (*As)[ASTR]) {
    #pragma unroll
    for (int e = tid; e < TM * KC; e += 256) {
        int i  = e & 127;
        int kk = e >> 7;
        int r  = Rb + i;
        int b  = r >> 4;
        int m  = r & 15;
        int kg = k0 + kk;
        float val = (kg < 1024) ? wr[(b << 14) + (kg << 4) + m]
                                : wi[(b << 14) + ((kg - 1024) << 4) + m];
        As[i][kk] = val;
    }
}

// One KC-deep chunk of WMMA work: wave 'wid' owns subtile row wid, 4 col subtiles.
__device__ __forceinline__ void wmma_chunk(const float (*As)[ASTR],
                                           const float (*Bs)[BSTR],
                                           int wid, int laneLo, int hsel,
                                           v8f acc[4]) {
    #pragma unroll
    for (int kk = 0; kk < KC; kk += 4) {
        const int kb = kk + hsel * 2;
        v2f a;
        a.x = As[wid * 16 + laneLo][kb];
        a.y = As[wid * 16 + laneLo][kb + 1];
        #pragma unroll
        for (int sc = 0; sc < 4; ++sc) {
            v2f bf;
            bf.x = Bs[kb][sc * 16 + laneLo];
            bf.y = Bs[kb + 1][sc * 16 + laneLo];
            acc[sc] = __builtin_amdgcn_wmma_f32_16x16x4_f32(
                false, a, false, bf, (short)0, acc[sc], false, false);
        }
    }
}

// ---------------------------------------------------------------------------
// Prep: S[b][0..1023] = cos(phi)/32, S[b][1024..2047] = sin(phi)/32
// ---------------------------------------------------------------------------
__global__ void prep_s_kernel(const float* __restrict__ phi, float* __restrict__ S) {
    int idx = blockIdx.x * blockDim.x + threadIdx.x;
    if (idx >= BB * LS) return;
    int b = idx >> 10;
    int k = idx & 1023;
    float sv, cv;
    sincosf(phi[idx], &sv, &cv);
    const float mag = 0.03125f;   // 1/sqrt(1024)
    S[b * 2048 + k]        = mag * cv;
    S[b * 2048 + 1024 + k] = mag * sv;
}

// ---------------------------------------------------------------------------
// TV[m][j] = v_M[:,m] . W1[4096+:,j] + b1[j]   (16 x 4096, tiny)
// ---------------------------------------------------------------------------
__global__ void tv_kernel(const float* __restrict__ vM, const float* __restrict__ W1,
                          const float* __restrict__ b1, float* __restrict__ TV) {
    int idx = blockIdx.x * blockDim.x + threadIdx.x;
    if (idx >= 16 * HH) return;
    int m = idx >> 12;
    int j = idx & 4095;
    float acc = b1[j];
    for (int k = 0; k < 512; ++k) {
        acc += vM[k * 16 + m] * W1[(size_t)(4096 + k) * HH + j];
    }
    TV[idx] = acc;
}

// ---------------------------------------------------------------------------
// TS = S @ W1[0:2048,:]    (256 x 4096), double-buffered WMMA GEMM
// ---------------------------------------------------------------------------
__global__ __launch_bounds__(256)
void ts_gemm_kernel(const float* __restrict__ S, const float* __restrict__ W1,
                    float* __restrict__ TS) {
    __shared__ float As[2][TM][ASTR];
    __shared__ float Bs[2][KC][BSTR];

    const int tid = threadIdx.x, wid = tid >> 5, lane = tid & 31;
    const int laneLo = lane & 15, hsel = lane >> 4;
    const int Rb = blockIdx.x * TM, Cb = blockIdx.y * TN;

    v8f acc[4] = {};
    constexpr int NT = KS / KC;

    load_A_dense(S, KS, Rb, 0, tid, As[0]);
    load_B_tile(W1, HH, 0, Cb, tid, Bs[0]);
    wait_async_zero();
    __syncthreads();

    for (int t = 0; t < NT; ++t) {
        const int cur = t & 1;
        if (t + 1 < NT) {
            load_A_dense(S, KS, Rb, (t + 1) * KC, tid, As[cur ^ 1]);
            load_B_tile(W1, HH, (t + 1) * KC, Cb, tid, Bs[cur ^ 1]);
        }
        wmma_chunk(As[cur], Bs[cur], wid, laneLo, hsel, acc);
        wait_async_zero();
        __syncthreads();
    }

    #pragma unroll
    for (int sc = 0; sc < 4; ++sc) {
        const int gc = Cb + sc * 16 + laneLo;
        #pragma unroll
        for (int i = 0; i < 8; ++i) {
            const int gr = Rb + wid * 16 + i + hsel * 8;
            TS[(size_t)gr * HH + gc] = acc[sc][i];
        }
    }
}

// ---------------------------------------------------------------------------
// HS[b][j] = sum_m relu( tw[b,m,j] + TS[b][j] + TV[m][j] )
// tw = [wr|wi] @ W1[2048:4096,:], rows r=b*16+m; m-reduction fused in epilogue.
// ---------------------------------------------------------------------------
__global__ __launch_bounds__(256)
void tw_gemm_kernel(const float* __restrict__ wr, const float* __restrict__ wi,
                    const float* __restrict__ W1c,   // = W1 + 2048*HH
                    const float* __restrict__ TS, const float* __restrict__ TV,
                    float* __restrict__ HS) {
    __shared__ float As[2][TM][ASTR];
    __shared__ float Bs[2][KC][BSTR];

    const int tid = threadIdx.x, wid = tid >> 5, lane = tid & 31;
    const int laneLo = lane & 15, hsel = lane >> 4;
    const int Rb = blockIdx.x * TM, Cb = blockIdx.y * TN;

    v8f acc[4] = {};
    constexpr int NT = KW / KC;

    load_A_w(wr, wi, Rb, 0, tid, As[0]);
    load_B_tile(W1c, HH, 0, Cb, tid, Bs[0]);
    wait_async_zero();
    __syncthreads();

    for (int t = 0; t < NT; ++t) {
        const int cur = t & 1;
        if (t + 1 < NT) {
            load_A_w(wr, wi, Rb, (t + 1) * KC, tid, As[cur ^ 1]);
            load_B_tile(W1c, HH, (t + 1) * KC, Cb, tid, Bs[cur ^ 1]);
        }
        wmma_chunk(As[cur], Bs[cur], wid, laneLo, hsel, acc);
        wait_async_zero();
        __syncthreads();
    }

    // bias terms + ReLU + in-register m-reduction; lanes 0..15 write HS.
    const int bIdx = (Rb >> 4) + wid;          // this wave's b-group
    #pragma unroll
    for (int sc = 0; sc < 4; ++sc) {
        const int gc = Cb + sc * 16 + laneLo;
        const float tsv = TS[(size_t)bIdx * HH + gc];
        float hp = 0.0f;
        #pragma unroll
        for (int i = 0; i < 8; ++i) {
            const int m = i + hsel * 8;
            hp += fmaxf(acc[sc][i] + tsv + TV[m * HH + gc], 0.0f);
        }
        hp += __shfl_xor(hp, 16, 32);          // m=0..7 half + m=8..15 half
        if (lane < 16) {
            HS[(size_t)bIdx * HH + gc] = hp;
        }
    }
}

// ---------------------------------------------------------------------------
// out[b][l] = phi[b][l] - ( HS[b] @ W2[:,l] + 16*b2[l] )
// ---------------------------------------------------------------------------
__global__ __launch_bounds__(256)
void out_gemm_kernel(const float* __restrict__ HS, const float* __restrict__ W2,
                     const float* __restrict__ b2, const float* __restrict__ phi,
                     float* __restrict__ out) {
    __shared__ float As[2][TM][ASTR];
    __shared__ float Bs[2][KC][BSTR];

    const int tid = threadIdx.x, wid = tid >> 5, lane = tid & 31;
    const int laneLo = lane & 15, hsel = lane >> 4;
    const int Rb = blockIdx.x * TM, Cb = blockIdx.y * TN;

    v8f acc[4] = {};
    constexpr int NT = HH / KC;

    load_A_dense(HS, HH, Rb, 0, tid, As[0]);
    load_B_tile(W2, LS, 0, Cb, tid, Bs[0]);
    wait_async_zero();
    __syncthreads();

    for (int t = 0; t < NT; ++t) {
        const int cur = t & 1;
        if (t + 1 < NT) {
            load_A_dense(HS, HH, Rb, (t + 1) * KC, tid, As[cur ^ 1]);
            load_B_tile(W2, LS, (t + 1) * KC, Cb, tid, Bs[cur ^ 1]);
        }
        wmma_chunk(As[cur], Bs[cur], wid, laneLo, hsel, acc);
        wait_async_zero();
        __syncthreads();
    }

    #pragma unroll
    for (int sc = 0; sc < 4; ++sc) {
        const int gc = Cb + sc * 16 + laneLo;
        const float bias16 = 16.0f * b2[gc];
        #pragma unroll
        for (int i = 0; i < 8; ++i) {
            const int gr = Rb + wid * 16 + i + hsel * 8;
            const size_t o = (size_t)gr * LS + gc;
            out[o] = phi[o] - (acc[sc][i] + bias16);
        }
    }
}

// ---------------------------------------------------------------------------
extern "C" void kernel_launch(void* const* d_in, const int* in_sizes, int n_in,
                              void* d_out, int out_size, void* d_ws, size_t ws_size,
                              hipStream_t stream) {
    const float* phi = (const float*)d_in[0];   // [256,1024]
    const float* wr  = (const float*)d_in[1];   // [256,1024,16]
    const float* wi  = (const float*)d_in[2];   // [256,1024,16]
    const float* vM  = (const float*)d_in[3];   // [512,16]
    const float* W1  = (const float*)d_in[4];   // [4608,4096]
    const float* b1  = (const float*)d_in[5];   // [4096]
    const float* W2  = (const float*)d_in[6];   // [4096,1024]
    const float* b2  = (const float*)d_in[7];   // [1024]
    float* out = (float*)d_out;                 // [256,1024]

    float* S  = (float*)d_ws;                   // [256][2048]  2 MB
    float* TS = S  + (size_t)BB * 2048;         // [256][4096]  4 MB
    float* TV = TS + (size_t)BB * HH;           // [16][4096]   256 KB
    float* HS = TV + (size_t)16 * HH;           // [256][4096]  4 MB

    prep_s_kernel<<<(BB * LS + 255) / 256, 256, 0, stream>>>(phi, S);

    tv_kernel<<<(16 * HH + 255) / 256, 256, 0, stream>>>(vM, W1, b1, TV);

    ts_gemm_kernel<<<dim3(BB / TM, HH / TN), 256, 0, stream>>>(S, W1, TS);

    tw_gemm_kernel<<<dim3(4096 / TM, HH / TN), 256, 0, stream>>>(
        wr, wi, W1 + (size_t)2048 * HH, TS, TV, HS);

    out_gemm_kernel<<<dim3(BB / TM, LS / TN), 256, 0, stream>>>(
        HS, W2, b2, phi, out);
}